// TransformerCritic_80942953661070
// MI455X (gfx1250) — compile-verified
//
#include <hip/hip_runtime.h>

// ---------------------------------------------------------------------------
// TransformerCritic forward for gfx1250 (CDNA5, wave32, WMMA)
// All dense layers + attention scores run through v_wmma_f32_16x16x4_f32.
// ---------------------------------------------------------------------------

typedef __attribute__((ext_vector_type(2))) float v2f;
typedef __attribute__((ext_vector_type(8))) float v8f;

#define BB 1024
#define NN 32
#define NAG (BB * NN)        // 32768 agents / rows
#define XK 96                // padded feature row: [conv32 | pose34 | act16 | pad14]
#define DE 128

__device__ __forceinline__ float leaky(float x) { return x > 0.f ? x : 0.01f * x; }

// ---------------------------------------------------------------------------
// Zero-pad a (rows x ksrc) row-major weight into (rows x 96)
// ---------------------------------------------------------------------------
__global__ void pad_w_kernel(const float* __restrict__ src, float* __restrict__ dst,
                             int rows, int ksrc) {
  int i = blockIdx.x * blockDim.x + threadIdx.x;
  if (i >= rows * XK) return;
  int r = i / XK, c = i % XK;
  dst[i] = (c < ksrc) ? src[r * ksrc + c] : 0.f;
}

// ---------------------------------------------------------------------------
// Conv1(3->16,3x3,VALID on 5x5) -> leaky -> Conv2(16->32,3x3,VALID) -> leaky,
// then assemble X row = [conv32 | gpos2 | onehot32 | actions16 | zeros14].
// One wave32 per agent, 8 agents per block.
// ---------------------------------------------------------------------------
#define AGB 8
__global__ void conv_assemble_kernel(const float* __restrict__ st,
                                     const float* __restrict__ gpos,
                                     const float* __restrict__ oneh,
                                     const float* __restrict__ act,
                                     const float* __restrict__ c1w,
                                     const float* __restrict__ c1b,
                                     const float* __restrict__ c2w,
                                     const float* __restrict__ c2b,
                                     float* __restrict__ X) {
  __shared__ float s_in[AGB][80];    // 3*5*5 input
  __shared__ float s_h1[AGB][144];   // 16*3*3 conv1 output
  const int wave = threadIdx.x >> 5;
  const int lane = threadIdx.x & 31;
  const int g = blockIdx.x * AGB + wave;           // agent id in [0, 32768)

  const float* in = st + (long)g * 75;
  for (int i = lane; i < 75; i += 32) s_in[wave][i] = in[i];
  __syncthreads();

  // conv1: 16 out-ch * 9 positions
  for (int idx = lane; idx < 144; idx += 32) {
    int oc = idx / 9, p = idx % 9, oy = p / 3, ox = p % 3;
    float sum = c1b[oc];
    #pragma unroll
    for (int ic = 0; ic < 3; ++ic)
      #pragma unroll
      for (int ky = 0; ky < 3; ++ky)
        #pragma unroll
        for (int kx = 0; kx < 3; ++kx)
          sum += s_in[wave][ic * 25 + (oy + ky) * 5 + (ox + kx)] *
                 c1w[oc * 27 + ic * 9 + ky * 3 + kx];
    s_h1[wave][idx] = leaky(sum);
  }
  __syncthreads();

  // conv2: one out-channel per lane (32 channels), 3x3 input fully consumed
  float sum = c2b[lane];
  #pragma unroll 4
  for (int ic = 0; ic < 16; ++ic)
    #pragma unroll
    for (int p = 0; p < 9; ++p)
      sum += s_h1[wave][ic * 9 + p] * c2w[lane * 144 + ic * 9 + p];
  float c2 = leaky(sum);

  float* xr = X + (long)g * XK;
  xr[lane] = c2;                                     // cols 0..31
  for (int c = lane + 32; c < XK; c += 32) {
    float v;
    if (c < 34)      v = gpos[(long)g * 2 + (c - 32)];
    else if (c < 66) v = oneh[(long)g * 32 + (c - 34)];
    else if (c < 82) v = act[(long)g * 16 + (c - 66)];
    else             v = 0.f;
    xr[c] = v;
  }
}

// ---------------------------------------------------------------------------
// Generic batched GEMM via V_WMMA_F32_16X16X4_F32.
//   Out[z][m][n] = scale * sum_k A[z][m][k] * W[z][n][k] + bias[n]   (opt leaky)
// One 16x16 tile per wave. Waves per block sweep N-tiles; tile guard is
// wave-uniform so EXEC stays all-ones inside WMMA.
// ---------------------------------------------------------------------------
__global__ void gemm_wmma_kernel(const float* __restrict__ A, int lda, long long sA,
                                 const float* __restrict__ W, int ldw, long long sW,
                                 const float* __restrict__ bias,
                                 float* __restrict__ Out, int ldo, long long sO,
                                 int K, int Ncols, float scale, int do_leaky) {
  const int lane = threadIdx.x & 31;
  const int wave = threadIdx.x >> 5;
  const int m0 = blockIdx.x * 16;
  const int n0 = (blockIdx.y * (blockDim.x >> 5) + wave) * 16;
  if (n0 >= Ncols) return;                  // uniform per wave

  const float* Ab = A + (long long)blockIdx.z * sA;
  const float* Wb = W + (long long)blockIdx.z * sW;
  float*       Ob = Out + (long long)blockIdx.z * sO;

  const int half = lane >> 4;               // selects K pair within the x4 step
  const int mr   = lane & 15;

  v8f acc = {};
  const float* arow = Ab + (long)(m0 + mr) * lda + 2 * half;
  const float* wrow = Wb + (long)(n0 + mr) * ldw + 2 * half;
  for (int k = 0; k < K; k += 4) {
    v2f a = *(const v2f*)(arow + k);        // A[m][klo], A[m][klo+1]
    v2f b = *(const v2f*)(wrow + k);        // W[n][klo], W[n][klo+1]  (= B^T row)
    acc = __builtin_amdgcn_wmma_f32_16x16x4_f32(false, a, false, b,
                                                (short)0, acc, false, false);
  }

  const int col = n0 + mr;
  const float bv = bias ? bias[col] : 0.f;
  #pragma unroll
  for (int gidx = 0; gidx < 8; ++gidx) {
    int row = m0 + gidx + 8 * half;         // C/D layout: VGPR g -> M=g / g+8
    float x = acc[gidx] * scale + bv;
    if (do_leaky) x = leaky(x);
    Ob[(long)row * ldo + col] = x;
  }
}

// ---------------------------------------------------------------------------
// Off-diagonal softmax + weighted aggregation. One wave32 per (b,n).
// Diagonal weight forced to 0 so a dense sum over all 32 m equals the
// off-diagonal gather. Writes `weight` in OFFDIAG order and node[:,128:256].
// ---------------------------------------------------------------------------
__global__ void attn_kernel(const float* __restrict__ score,
                            const float* __restrict__ vals,
                            float* __restrict__ node,
                            float* __restrict__ wout) {
  const int wave = threadIdx.x >> 5;
  const int lane = threadIdx.x & 31;       // lane == m (peer agent)
  const int g = blockIdx.x * 8 + wave;     // row id = b*32 + n
  const int b = g >> 5;
  const int n = g & 31;

  float s  = score[(long)g * 32 + lane];
  float se = (lane == n) ? -3.4e38f : s;
  float mx = se;
  #pragma unroll
  for (int off = 16; off; off >>= 1) mx = fmaxf(mx, __shfl_xor(mx, off, 32));
  float e = (lane == n) ? 0.f : __expf(se - mx);
  float tot = e;
  #pragma unroll
  for (int off = 16; off; off >>= 1) tot += __shfl_xor(tot, off, 32);
  float w = e / tot;

  if (lane != n) {
    int c = (lane < n) ? lane : lane - 1;  // OFFDIAG position
    wout[(long)g * 31 + c] = w;
  }

  // agg[k] = sum_m w[m] * vals[b,m,k]; each lane owns 4 of the 128 dims
  const float* vb = vals + (long)b * 32 * 128;
  const int k = lane * 4;
  float4 acc = make_float4(0.f, 0.f, 0.f, 0.f);
  #pragma unroll 8
  for (int m = 0; m < 32; ++m) {
    float wm = __shfl(w, m, 32);
    float4 vv = *(const float4*)(vb + m * 128 + k);
    acc.x += wm * vv.x; acc.y += wm * vv.y;
    acc.z += wm * vv.z; acc.w += wm * vv.w;
  }
  *(float4*)(node + (long)g * 256 + 128 + k) = acc;
}

// ---------------------------------------------------------------------------
// Host-side orchestration
// ---------------------------------------------------------------------------
extern "C" void kernel_launch(void* const* d_in, const int* in_sizes, int n_in,
                              void* d_out, int out_size, void* d_ws, size_t ws_size,
                              hipStream_t stream) {
  const float* agent_states = (const float*)d_in[0];
  const float* gpos         = (const float*)d_in[1];
  const float* onehot       = (const float*)d_in[2];
  // d_in[3] = policies (unused by reference)
  const float* actions      = (const float*)d_in[4];
  const float* c1w = (const float*)d_in[5];
  const float* c1b = (const float*)d_in[6];
  const float* c2w = (const float*)d_in[7];
  const float* c2b = (const float*)d_in[8];
  const float* sa_w = (const float*)d_in[9];
  const float* sa_b = (const float*)d_in[10];
  const float* s_w  = (const float*)d_in[11];
  const float* s_b  = (const float*)d_in[12];
  const float* k_w  = (const float*)d_in[13];
  const float* k_b  = (const float*)d_in[14];
  const float* q_w  = (const float*)d_in[15];
  const float* q_b  = (const float*)d_in[16];
  const float* v_w  = (const float*)d_in[17];
  const float* v_b  = (const float*)d_in[18];
  const float* sq_w = (const float*)d_in[19];
  const float* sq_b = (const float*)d_in[20];
  const float* f1_w = (const float*)d_in[21];
  const float* f1_b = (const float*)d_in[22];
  const float* f2_w = (const float*)d_in[23];
  const float* f2_b = (const float*)d_in[24];

  float* ws = (float*)d_ws;
  // workspace layout (floats)
  float* X      = ws;                        // 32768*96   (reused later for h1: 32768*64)
  float* Wsp    = X + (long)NAG * XK;        // s_w_pad
  float* Wsap   = Wsp + 128 * XK;            // sa_w_pad
  float* Wsqp   = Wsap + 128 * XK;           // sq_w_pad
  float* se     = Wsqp + 128 * XK;           // 32768*128 (reused later for score: 1024*32*32)
  float* sa     = se + (long)NAG * DE;       // 32768*128
  float* qbuf   = sa + (long)NAG * DE;       // 32768*128
  float* kbuf   = qbuf + (long)NAG * DE;     // 32768*128
  float* vbuf   = kbuf + (long)NAG * DE;     // 32768*128
  float* node   = vbuf + (long)NAG * DE;     // 32768*256
  float* score  = se;                        // overlay: se dead after q GEMM
  float* h1     = X;                         // overlay: X dead after se/sa/sq GEMMs

  float* value_out  = (float*)d_out;                 // 32768*16
  float* weight_out = (float*)d_out + (long)NAG * 16; // 32768*31

  const float inv_sqrt_d = 0.08838834764831845f;     // 1/sqrt(128)

  // 1. pad weights to K=96
  pad_w_kernel<<<48, 256, 0, stream>>>(s_w,  Wsp,  128, 66);
  pad_w_kernel<<<48, 256, 0, stream>>>(sa_w, Wsap, 128, 82);
  pad_w_kernel<<<48, 256, 0, stream>>>(sq_w, Wsqp, 128, 66);

  // 2. conv + assemble X
  conv_assemble_kernel<<<NAG / AGB, 32 * AGB, 0, stream>>>(
      agent_states, gpos, onehot, actions, c1w, c1b, c2w, c2b, X);

  // 3. se / sa / sq  (K=96 padded), sq writes directly into node[:, :128]
  dim3 gA(NAG / 16, 2, 1);   // 128 N-cols, 4 waves/block
  gemm_wmma_kernel<<<gA, 128, 0, stream>>>(X, XK, 0, Wsp,  XK, 0, s_b,
                                           se, DE, 0, XK, DE, 1.f, 1);
  gemm_wmma_kernel<<<gA, 128, 0, stream>>>(X, XK, 0, Wsap, XK, 0, sa_b,
                                           sa, DE, 0, XK, DE, 1.f, 1);
  gemm_wmma_kernel<<<gA, 128, 0, stream>>>(X, XK, 0, Wsqp, XK, 0, sq_b,
                                           node, 256, 0, XK, DE, 1.f, 1);

  // 4. queries(se), keys(sa), vals(sa): 128 -> 128
  gemm_wmma_kernel<<<gA, 128, 0, stream>>>(se, DE, 0, q_w, DE, 0, q_b,
                                           qbuf, DE, 0, DE, DE, 1.f, 0);
  gemm_wmma_kernel<<<gA, 128, 0, stream>>>(sa, DE, 0, k_w, DE, 0, k_b,
                                           kbuf, DE, 0, DE, DE, 1.f, 0);
  gemm_wmma_kernel<<<gA, 128, 0, stream>>>(sa, DE, 0, v_w, DE, 0, v_b,
                                           vbuf, DE, 0, DE, DE, 1.f, 0);

  // 5. score[b] = Q_b(32x128) @ K_b(32x128)^T / sqrt(128)  — batched over b
  dim3 gS(2, 1, BB);         // 2 M-tiles; waves cover 2 N-tiles (32 cols)
  gemm_wmma_kernel<<<gS, 128, 0, stream>>>(qbuf, DE, (long long)NN * DE,
                                           kbuf, DE, (long long)NN * DE,
                                           nullptr,
                                           score, NN, (long long)NN * NN,
                                           DE, NN, inv_sqrt_d, 0);

  // 6. softmax (off-diagonal) + aggregation -> weight_out, node[:,128:]
  attn_kernel<<<NAG / 8, 256, 0, stream>>>(score, vbuf, node, weight_out);

  // 7. head: f1 (256->64, leaky) then f2 (64->16) -> value_out
  dim3 gF1(NAG / 16, 1, 1);  // 64 cols, 4 waves/block
  gemm_wmma_kernel<<<gF1, 128, 0, stream>>>(node, 256, 0, f1_w, 256, 0, f1_b,
                                            h1, 64, 0, 256, 64, 1.f, 1);
  dim3 gF2(NAG / 16, 1, 1);  // 16 cols, wave 0 only
  gemm_wmma_kernel<<<gF2, 128, 0, stream>>>(h1, 64, 0, f2_w, 64, 0, f2_b,
                                            value_out, 16, 0, 64, 16, 1.f, 0);
}